// STGCN_LSTM_23991687315688
// MI455X (gfx1250) — compile-verified
//
#include <hip/hip_runtime.h>
#include <hip/hip_bf16.h>

#define NN     20000
#define EE     320000
#define F_NODE 64
#define F_EDGE 16
#define NH     4
#define HD     32
#define DD     128

typedef __attribute__((ext_vector_type(2))) float v2f;
typedef __attribute__((ext_vector_type(8))) float v8f;

__device__ __forceinline__ unsigned f2u(float f) {
    unsigned u = __float_as_uint(f);
    return (u & 0x80000000u) ? ~u : (u | 0x80000000u);
}
__device__ __forceinline__ float u2f(unsigned u) {
    unsigned v = (u & 0x80000000u) ? (u & 0x7fffffffu) : ~u;
    return __uint_as_float(v);
}
__device__ __forceinline__ float sigm(float x) { return 1.0f / (1.0f + __expf(-x)); }

// ---------------------------------------------------------------------------
// WMMA f32 GEMM: hp[N x 128] = X[N x Kdim] @ W[Kdim x 128]
// 1 wave -> one 16x16 output tile; block = 256 threads = 8 waves = 8 col tiles
// A frag lane L, vgpr v : K = kbase + 2*(L/16) + v, M = L%16
// B frag lane L, vgpr v : K = kbase + 2*(L/16) + v, N = L%16
// C/D   lane L, vgpr v  : M = v + 8*(L/16),         N = L%16
// ---------------------------------------------------------------------------
__global__ __launch_bounds__(256) void gat_gemm_wmma(
    const float* __restrict__ X, const float* __restrict__ W,
    float* __restrict__ hp, int Kdim) {
    const int lane  = threadIdx.x & 31;
    const int wave  = threadIdx.x >> 5;       // col tile 0..7
    const int rowT  = blockIdx.x;             // row tile 0..1249
    const int m     = lane & 15;
    const int khalf = (lane >> 4) * 2;        // 0 or 2
    const int col   = wave * 16 + m;

    const float* xrow = X + (size_t)(rowT * 16 + m) * Kdim;
    v8f c = {};
    for (int k = 0; k < Kdim; k += 4) {
        v2f a, b;
        a.x = xrow[k + khalf + 0];
        a.y = xrow[k + khalf + 1];
        b.x = W[(size_t)(k + khalf + 0) * DD + col];
        b.y = W[(size_t)(k + khalf + 1) * DD + col];
        c = __builtin_amdgcn_wmma_f32_16x16x4_f32(false, a, false, b,
                                                  (short)0, c, false, false);
    }
    float* outp = hp + (size_t)(rowT * 16 + 8 * (lane >> 4)) * DD + col;
#pragma unroll
    for (int v = 0; v < 8; ++v) outp[(size_t)v * DD] = c[v];
}

// M[k,h] = sum_d We[k, h*32+d] * ae[h,d]   (collapses the E x 128 edge GEMM)
__global__ void precomp_edge_mat(const float* __restrict__ We,
                                 const float* __restrict__ ae,
                                 float* __restrict__ M) {
    int t = threadIdx.x;
    if (t < F_EDGE * NH) {
        int k = t >> 2, h = t & 3;
        float s = 0.f;
#pragma unroll 8
        for (int j = 0; j < HD; ++j) s += We[k * DD + h * HD + j] * ae[h * HD + j];
        M[k * NH + h] = s;
    }
}

// per-node attention logit halves: asrc[n,h], adst[n,h]
__global__ __launch_bounds__(256) void node_alpha(
    const float* __restrict__ hp, const float* __restrict__ a_s,
    const float* __restrict__ a_d, float* __restrict__ asrc,
    float* __restrict__ adst) {
    int n = blockIdx.x * blockDim.x + threadIdx.x;
    if (n >= NN) return;
    const float* r = hp + (size_t)n * DD;
#pragma unroll
    for (int h = 0; h < NH; ++h) {
        float s = 0.f, d = 0.f;
#pragma unroll 8
        for (int j = 0; j < HD; ++j) {
            float v = r[h * HD + j];
            s += v * a_s[h * HD + j];
            d += v * a_d[h * HD + j];
        }
        asrc[n * NH + h] = s;
        adst[n * NH + h] = d;
    }
}

__global__ __launch_bounds__(256) void init_buffers(
    float* __restrict__ agg, unsigned* __restrict__ m_u,
    float* __restrict__ den) {
    int i = blockIdx.x * blockDim.x + threadIdx.x;
    if (i < NN * DD) agg[i] = 0.f;
    if (i < NN * NH) { m_u[i] = 0u; den[i] = 0.f; }
}

// pass 1: logits + leaky_relu + scatter-max
__global__ __launch_bounds__(256) void edge_logits(
    const int* __restrict__ ei, const float* __restrict__ ea,
    const float* __restrict__ asrc, const float* __restrict__ adst,
    const float* __restrict__ M, float* __restrict__ al,
    unsigned* __restrict__ m_u) {
    int e = blockIdx.x * blockDim.x + threadIdx.x;
    if (e >= EE) return;
    int s = ei[e], d = ei[EE + e];
    float f[F_EDGE];
#pragma unroll
    for (int k = 0; k < F_EDGE; ++k) f[k] = ea[(size_t)e * F_EDGE + k];
#pragma unroll
    for (int h = 0; h < NH; ++h) {
        float le = 0.f;
#pragma unroll
        for (int k = 0; k < F_EDGE; ++k) le += f[k] * M[k * NH + h];
        float a = asrc[s * NH + h] + adst[d * NH + h] + le;
        a = (a >= 0.f) ? a : 0.2f * a;                 // leaky_relu 0.2
        al[(size_t)e * NH + h] = a;
        atomicMax(&m_u[d * NH + h], f2u(a));
    }
}

// pass 2: exp(al - max) + scatter-sum denominator
__global__ __launch_bounds__(256) void edge_exp(
    const int* __restrict__ ei, float* __restrict__ al,
    const unsigned* __restrict__ m_u, float* __restrict__ den) {
    int e = blockIdx.x * blockDim.x + threadIdx.x;
    if (e >= EE) return;
    int d = ei[EE + e];
#pragma unroll
    for (int h = 0; h < NH; ++h) {
        float mx = u2f(m_u[d * NH + h]);
        float ex = __expf(al[(size_t)e * NH + h] - mx);
        al[(size_t)e * NH + h] = ex;
        atomicAdd(&den[d * NH + h], ex);
    }
}

// pass 3: agg[dst] += hp[src] * alpha    (one thread per (edge, head))
__global__ __launch_bounds__(256) void edge_msg(
    const int* __restrict__ ei, const float* __restrict__ al,
    const float* __restrict__ den, const float* __restrict__ hp,
    float* __restrict__ agg) {
    int t = blockIdx.x * blockDim.x + threadIdx.x;
    if (t >= EE * NH) return;
    int e = t >> 2, h = t & 3;
    int s = ei[e], d = ei[EE + e];
    float alpha = al[(size_t)e * NH + h] / (den[d * NH + h] + 1e-16f);
    const float4* sp = (const float4*)(hp + (size_t)s * DD + h * HD);
    float* dp = agg + (size_t)d * DD + h * HD;
    __builtin_prefetch(sp, 0, 0);   // global_prefetch_b8
#pragma unroll
    for (int q = 0; q < HD / 4; ++q) {
        float4 v = sp[q];
        atomicAdd(dp + q * 4 + 0, v.x * alpha);
        atomicAdd(dp + q * 4 + 1, v.y * alpha);
        atomicAdd(dp + q * 4 + 2, v.z * alpha);
        atomicAdd(dp + q * 4 + 3, v.w * alpha);
    }
}

// epilogue: h_out = elu(agg + b)
__global__ __launch_bounds__(256) void node_elu(
    const float* __restrict__ agg, const float* __restrict__ b,
    float* __restrict__ out) {
    int i = blockIdx.x * blockDim.x + threadIdx.x;
    if (i >= NN * DD) return;
    float v = agg[i] + b[i & (DD - 1)];
    out[i] = (v > 0.f) ? v : (__expf(v) - 1.f);
}

// ---------------------------------------------------------------------------
// Sequential LSTM over the node axis (20000 dependent steps), single block.
// 512 threads: thread (r = tid/4, p = tid%4) keeps Wih[r, p*32..] (32 regs)
// and Whh[r, p*8..] (8 regs); 4-way partial reduction through LDS.
// Tail: out64[f] = fcb[f] + fcW[f,:] . h_T
// ---------------------------------------------------------------------------
__global__ __launch_bounds__(512) void lstm_seq(
    const float* __restrict__ h2, const float* __restrict__ Wih,
    const float* __restrict__ Whh, const float* __restrict__ bih,
    const float* __restrict__ bhh, const float* __restrict__ fcW,
    const float* __restrict__ fcb, float* __restrict__ out64) {
    __shared__ float part[512];
    __shared__ float bias_s[4 * HD];
    __shared__ float h_s[HD];

    const int tid = threadIdx.x;
    const int r = tid >> 2;      // gate row 0..127
    const int p = tid & 3;       // partial slice 0..3

    float wih[32], whh[8];
#pragma unroll
    for (int j = 0; j < 32; ++j) wih[j] = Wih[r * DD + p * 32 + j];
#pragma unroll
    for (int j = 0; j < 8; ++j) whh[j] = Whh[r * HD + p * 8 + j];

    if (tid < 4 * HD) bias_s[tid] = bih[tid] + bhh[tid];
    if (tid < HD) h_s[tid] = 0.f;
    float c = 0.f;               // valid in threads 0..31
    __syncthreads();

    for (int t = 0; t < NN; ++t) {
        const float* xt = h2 + (size_t)t * DD + p * 32;
        float acc = 0.f;
#pragma unroll 8
        for (int j = 0; j < 32; ++j) acc += wih[j] * xt[j];
#pragma unroll
        for (int j = 0; j < 8; ++j) acc += whh[j] * h_s[p * 8 + j];
        part[tid] = acc;
        __syncthreads();
        if (tid < HD) {
            int b0 = 4 * tid, b1 = 4 * (HD + tid), b2 = 4 * (2 * HD + tid),
                b3 = 4 * (3 * HD + tid);
            float gi = bias_s[tid]          + part[b0] + part[b0 + 1] + part[b0 + 2] + part[b0 + 3];
            float gf = bias_s[HD + tid]     + part[b1] + part[b1 + 1] + part[b1 + 2] + part[b1 + 3];
            float gg = bias_s[2 * HD + tid] + part[b2] + part[b2 + 1] + part[b2 + 2] + part[b2 + 3];
            float go = bias_s[3 * HD + tid] + part[b3] + part[b3 + 1] + part[b3 + 2] + part[b3 + 3];
            c = sigm(gf) * c + sigm(gi) * tanhf(gg);
            h_s[tid] = sigm(go) * tanhf(c);
        }
        __syncthreads();
    }

    if (tid < F_NODE) {
        float s = fcb[tid];
#pragma unroll 8
        for (int j = 0; j < HD; ++j) s += fcW[tid * HD + j] * h_s[j];
        out64[tid] = s;
    }
}

// final: out[n] = projb[n] + projW[n,:] . out64
__global__ __launch_bounds__(256) void final_proj(
    const float* __restrict__ projW, const float* __restrict__ projb,
    const float* __restrict__ out64, float* __restrict__ out) {
    int n = blockIdx.x * blockDim.x + threadIdx.x;
    if (n >= NN) return;
    float s = projb[n];
#pragma unroll 8
    for (int f = 0; f < F_NODE; ++f) s += projW[(size_t)n * F_NODE + f] * out64[f];
    out[n] = s;
}

// ---------------------------------------------------------------------------
extern "C" void kernel_launch(void* const* d_in, const int* in_sizes, int n_in,
                              void* d_out, int out_size, void* d_ws, size_t ws_size,
                              hipStream_t stream) {
    const float* x    = (const float*)d_in[0];
    const int*   ei   = (const int*)  d_in[1];
    const float* ea   = (const float*)d_in[2];
    const float* W0   = (const float*)d_in[4];
    const float* We0  = (const float*)d_in[5];
    const float* as0  = (const float*)d_in[6];
    const float* ad0  = (const float*)d_in[7];
    const float* ae0  = (const float*)d_in[8];
    const float* b0   = (const float*)d_in[9];
    const float* W1   = (const float*)d_in[10];
    const float* We1  = (const float*)d_in[11];
    const float* as1  = (const float*)d_in[12];
    const float* ad1  = (const float*)d_in[13];
    const float* ae1  = (const float*)d_in[14];
    const float* b1   = (const float*)d_in[15];
    const float* Wih  = (const float*)d_in[16];
    const float* Whh  = (const float*)d_in[17];
    const float* bih  = (const float*)d_in[18];
    const float* bhh  = (const float*)d_in[19];
    const float* fcW  = (const float*)d_in[20];
    const float* fcb  = (const float*)d_in[21];
    const float* prjW = (const float*)d_in[22];
    const float* prjb = (const float*)d_in[23];
    float* outp = (float*)d_out;

    char* ws = (char*)d_ws;
    size_t off = 0;
    float*    hp   = (float*)(ws + off); off += (size_t)NN * DD * 4;   // 10.24 MB
    float*    h1   = (float*)(ws + off); off += (size_t)NN * DD * 4;   // 10.24 MB
    float*    agg  = (float*)(ws + off); off += (size_t)NN * DD * 4;   // 10.24 MB
    float*    exb  = (float*)(ws + off); off += (size_t)EE * NH * 4;   // 5.12 MB
    float*    asrc = (float*)(ws + off); off += (size_t)NN * NH * 4;
    float*    adst = (float*)(ws + off); off += (size_t)NN * NH * 4;
    unsigned* m_u  = (unsigned*)(ws + off); off += (size_t)NN * NH * 4;
    float*    den  = (float*)(ws + off); off += (size_t)NN * NH * 4;
    float*    Mm   = (float*)(ws + off); off += 512;
    float*    o64  = (float*)(ws + off); off += 256;

    const int gN   = (NN + 255) / 256;          // 79
    const int gND  = (NN * DD + 255) / 256;     // 10000
    const int gE   = (EE + 255) / 256;          // 1250
    const int gEH  = (EE * NH + 255) / 256;     // 5000

    // ---- GAT layer 0 ----
    precomp_edge_mat<<<1, 64, 0, stream>>>(We0, ae0, Mm);
    gat_gemm_wmma<<<NN / 16, 256, 0, stream>>>(x, W0, hp, F_NODE);
    node_alpha<<<gN, 256, 0, stream>>>(hp, as0, ad0, asrc, adst);
    init_buffers<<<gND, 256, 0, stream>>>(agg, m_u, den);
    edge_logits<<<gE, 256, 0, stream>>>(ei, ea, asrc, adst, Mm, exb, m_u);
    edge_exp<<<gE, 256, 0, stream>>>(ei, exb, m_u, den);
    edge_msg<<<gEH, 256, 0, stream>>>(ei, exb, den, hp, agg);
    node_elu<<<gND, 256, 0, stream>>>(agg, b0, h1);

    // ---- GAT layer 1 ----
    precomp_edge_mat<<<1, 64, 0, stream>>>(We1, ae1, Mm);
    gat_gemm_wmma<<<NN / 16, 256, 0, stream>>>(h1, W1, hp, DD);
    node_alpha<<<gN, 256, 0, stream>>>(hp, as1, ad1, asrc, adst);
    init_buffers<<<gND, 256, 0, stream>>>(agg, m_u, den);
    edge_logits<<<gE, 256, 0, stream>>>(ei, ea, asrc, adst, Mm, exb, m_u);
    edge_exp<<<gE, 256, 0, stream>>>(ei, exb, m_u, den);
    edge_msg<<<gEH, 256, 0, stream>>>(ei, exb, den, hp, agg);
    node_elu<<<gND, 256, 0, stream>>>(agg, b1, h1);   // h2 reuses h1 buffer

    // ---- LSTM over node axis + fc ----
    lstm_seq<<<1, 512, 0, stream>>>(h1, Wih, Whh, bih, bhh, fcW, fcb, o64);

    // ---- final projection to [1, N] ----
    final_proj<<<gN, 256, 0, stream>>>(prjW, prjb, o64, outp);
}